// HGCN_76063870812433
// MI455X (gfx1250) — compile-verified
//
#include <hip/hip_runtime.h>

typedef __attribute__((ext_vector_type(2))) float v2f;
typedef __attribute__((ext_vector_type(8))) float v8f;

// -------------------------------------------------------------------------
// zero-fill (accumulators + degree arrays must start at 0 every call)
// -------------------------------------------------------------------------
__global__ void zero_f32_kernel(float* __restrict__ p, long n) {
  long i = (long)blockIdx.x * blockDim.x + threadIdx.x;
  long stride = (long)gridDim.x * blockDim.x;
  for (; i < n; i += stride) p[i] = 0.0f;
}

// -------------------------------------------------------------------------
// Edge scatter: one wave32 per edge. Each lane moves 4 consecutive floats of
// the 128-float source row (float4 load -> 512B coalesced per wave) and
// atomically accumulates into the destination row (L2-resident, 25.6MB).
// Lane 0 also counts the in-degree.
// -------------------------------------------------------------------------
__global__ __launch_bounds__(256) void scatter_mean_kernel(
    const float* __restrict__ hsrc, const int* __restrict__ src,
    const int* __restrict__ dst, int E, float* __restrict__ acc,
    float* __restrict__ deg) {
  int lane = threadIdx.x & 31;
  long wid = ((long)blockIdx.x * blockDim.x + threadIdx.x) >> 5;
  long nw  = ((long)gridDim.x * blockDim.x) >> 5;
  for (long e = wid; e < (long)E; e += nw) {
    int s = src[e];
    int d = dst[e];
    const float4* srow = (const float4*)(hsrc + (long)s * 128);
    float4 v = srow[lane];
    float* drow = acc + (long)d * 128 + lane * 4;
    unsafeAtomicAdd(drow + 0, v.x);
    unsafeAtomicAdd(drow + 1, v.y);
    unsafeAtomicAdd(drow + 2, v.z);
    unsafeAtomicAdd(drow + 3, v.w);
    if (lane == 0) unsafeAtomicAdd(deg + d, 1.0f);
  }
}

// -------------------------------------------------------------------------
// SAGE transform with fp32 WMMA (V_WMMA_F32_16X16X4_F32):
//   out[m,n] (= or +=) relu( H[m,:]@Ws[:,n] + (N[m,:]/max(deg[m],1))@Wn[:,n]
//                            + bias[n] )
// Block = 256 threads = 8 waves. Block handles rows [16*bx,16*bx+16),
// wave w handles cols [16w,16w+16). K=128 stepped by 4, two WMMAs per step
// (self path + neighbor path) accumulating into one 16x16 fp32 tile.
//
// VGPR layouts per CDNA5 ISA 7.12.2:
//   A 16x4 : lane l -> row l%16, holds K = 2*(l/16)+{0,1}   (contig float2)
//   B 4x16 : lane l -> col l%16, holds K = 2*(l/16)+{0,1}
//   C 16x16: vgpr v, lane l -> row v + 8*(l/16), col l%16
//
// ADD is a template parameter so the accumulate-into-out variant carries no
// runtime branch; the in-bounds check is one uniform block-level branch.
// -------------------------------------------------------------------------
template <int ADD>
__global__ __launch_bounds__(256) void sage_wmma_kernel(
    const float* __restrict__ H, const float* __restrict__ N,
    const float* __restrict__ deg, const float* __restrict__ Ws,
    const float* __restrict__ Wn, const float* __restrict__ bias,
    float* __restrict__ out, int ns) {
  const int wave = threadIdx.x >> 5;
  const int lane = threadIdx.x & 31;
  const int m0   = blockIdx.x * 16;
  const int cb   = wave * 16;           // 8 waves cover all 128 output cols
  const int rloc = lane & 15;           // A row within tile / B,C column
  const int kp   = (lane >> 4) << 1;    // K sub-offset: 0 or 2
  const int col  = cb + rloc;

  const int rowm = min(m0 + rloc, ns - 1);
  const float* hrow = H + (long)rowm * 128;
  const float* nrow = N + (long)rowm * 128;
  const float scale = 1.0f / fmaxf(deg[rowm], 1.0f);

  v8f acc = {};
#pragma unroll 4
  for (int k0 = 0; k0 < 128; k0 += 4) {
    v2f a, b;
    // self path: H @ Ws
    a.x = hrow[k0 + kp];
    a.y = hrow[k0 + kp + 1];
    b.x = Ws[(long)(k0 + kp) * 128 + col];
    b.y = Ws[(long)(k0 + kp + 1) * 128 + col];
    acc = __builtin_amdgcn_wmma_f32_16x16x4_f32(false, a, false, b,
                                                (short)0, acc, false, false);
    // neighbor path: (N/deg) @ Wn   (mean normalization folded into A-load)
    a.x = nrow[k0 + kp] * scale;
    a.y = nrow[k0 + kp + 1] * scale;
    b.x = Wn[(long)(k0 + kp) * 128 + col];
    b.y = Wn[(long)(k0 + kp + 1) * 128 + col];
    acc = __builtin_amdgcn_wmma_f32_16x16x4_f32(false, a, false, b,
                                                (short)0, acc, false, false);
  }

  const float bcol = bias[col];
  const int rbase  = m0 + ((lane >> 4) << 3);   // row of acc[0] for this lane

  if (m0 + 16 <= ns) {
    // full tile: uniform branch, no per-element guards
    float* o = out + (long)rbase * 128 + col;
#pragma unroll
    for (int v = 0; v < 8; ++v) {
      float val = fmaxf(acc[v] + bcol, 0.0f);
      if (ADD) val += o[(long)v * 128];
      o[(long)v * 128] = val;
    }
  } else {
    // remainder tile (unused for NS % 16 == 0, kept for generality)
#pragma unroll
    for (int v = 0; v < 8; ++v) {
      int r = rbase + v;
      if (r < ns) {
        float val = fmaxf(acc[v] + bcol, 0.0f);
        float* o = out + (long)r * 128 + col;
        if (ADD) val += *o;
        *o = val;
      }
    }
  }
}

// -------------------------------------------------------------------------
// Host orchestration
// inputs: 0 h_station[NS,128] 1 h_feature[NF,128] 2 Wself[3,2,128,128]
//         3 Wneigh[3,2,128,128] 4 bias[3,2,128]
//         5 hf_src[E1] 6 hf_dst[E1] 7 tt_src[E2] 8 tt_dst[E2]
// -------------------------------------------------------------------------
extern "C" void kernel_launch(void* const* d_in, const int* in_sizes, int n_in,
                              void* d_out, int out_size, void* d_ws,
                              size_t ws_size, hipStream_t stream) {
  const float* h_station = (const float*)d_in[0];
  const float* h_feature = (const float*)d_in[1];
  const float* Wself     = (const float*)d_in[2];
  const float* Wneigh    = (const float*)d_in[3];
  const float* bias      = (const float*)d_in[4];
  const int* hf_src      = (const int*)d_in[5];
  const int* hf_dst      = (const int*)d_in[6];
  const int* tt_src      = (const int*)d_in[7];
  const int* tt_dst      = (const int*)d_in[8];
  const int NS = in_sizes[0] / 128;
  const int E1 = in_sizes[5];
  const int E2 = in_sizes[7];
  float* out = (float*)d_out;

  // workspace: [accA | accB | h1 | degA | degB]  (~77 MB for NS=50000)
  float* accA = (float*)d_ws;              // n_hf accumulator; reused as h2
  float* accB = accA + (long)NS * 128;     // n_tt accumulator
  float* h1   = accB + (long)NS * 128;     // layer-1 output
  float* degA = h1 + (long)NS * 128;
  float* degB = degA + NS;

  auto woff = [](int l, int r) { return (long)(l * 2 + r) * 128 * 128; };
  auto boff = [](int l, int r) { return (long)(l * 2 + r) * 128; };

  const int zgrid = 2048;                  // zero-fill blocks
  const int sgrid = 2048;                  // 16384 waves grid-striding edges
  const int ggrid = (NS + 15) / 16;        // 3125 tile-row blocks

  // ---- layer 1: two relations into stations --------------------------------
  zero_f32_kernel<<<zgrid, 256, 0, stream>>>(accA, 2L * NS * 128);
  zero_f32_kernel<<<64, 256, 0, stream>>>(degA, 2L * NS);
  scatter_mean_kernel<<<sgrid, 256, 0, stream>>>(h_feature, hf_src, hf_dst, E1,
                                                 accA, degA);
  scatter_mean_kernel<<<sgrid, 256, 0, stream>>>(h_station, tt_src, tt_dst, E2,
                                                 accB, degB);
  sage_wmma_kernel<0><<<ggrid, 256, 0, stream>>>(
      h_station, accA, degA, Wself + woff(0, 0), Wneigh + woff(0, 0),
      bias + boff(0, 0), h1, NS);
  sage_wmma_kernel<1><<<ggrid, 256, 0, stream>>>(
      h_station, accB, degB, Wself + woff(0, 1), Wneigh + woff(0, 1),
      bias + boff(0, 1), h1, NS);

  // ---- layer 2 -------------------------------------------------------------
  zero_f32_kernel<<<zgrid, 256, 0, stream>>>(accB, (long)NS * 128);
  zero_f32_kernel<<<64, 256, 0, stream>>>(degB, (long)NS);
  scatter_mean_kernel<<<sgrid, 256, 0, stream>>>(h1, tt_src, tt_dst, E2, accB,
                                                 degB);
  sage_wmma_kernel<0><<<ggrid, 256, 0, stream>>>(
      h1, accB, degB, Wself + woff(1, 1), Wneigh + woff(1, 1),
      bias + boff(1, 1), accA, NS);      // h2 := accA (free after layer 1)

  // ---- layer 3 -------------------------------------------------------------
  zero_f32_kernel<<<zgrid, 256, 0, stream>>>(accB, (long)NS * 128);
  zero_f32_kernel<<<64, 256, 0, stream>>>(degB, (long)NS);
  scatter_mean_kernel<<<sgrid, 256, 0, stream>>>(accA, tt_src, tt_dst, E2, accB,
                                                 degB);
  sage_wmma_kernel<0><<<ggrid, 256, 0, stream>>>(
      accA, accB, degB, Wself + woff(2, 1), Wneigh + woff(2, 1),
      bias + boff(2, 1), out, NS);
}